// EdgeFeature_8400956031125
// MI455X (gfx1250) — compile-verified
//
#include <hip/hip_runtime.h>
#include <math.h>

typedef __attribute__((ext_vector_type(2))) float v2f;
typedef __attribute__((ext_vector_type(4))) float v4f;
typedef __attribute__((ext_vector_type(8))) float v8f;

#define NRBF 8
#define NSPH 16
#define INV_SQRT_4PI 0.28209479177387814f

// One wave handles 32 edges:
//   phase 1: per-lane scalar feature compute -> LDS
//   phase 2: 16x V_WMMA_F32_16X16X4_F32 (two edges per WMMA, K=2 of 4 used),
//            D tile scattered to padded LDS staging, then fully coalesced
//            global_store_b128 NT stores (store bandwidth is the roofline).
__global__ __launch_bounds__(256) void edge_feature_kernel(
    const int* __restrict__ zmap, const int* __restrict__ idx_i,
    const int* __restrict__ idx_j, const float* __restrict__ r_ij,
    const float* __restrict__ embed, float* __restrict__ out, int P)
{
    __shared__ float s_sph[8][32][NSPH];                 // sph * (x_i*x_j/sqrt(4pi))
    __shared__ float s_rbf[8][32][NRBF];
    __shared__ __align__(16) float s_stage[8][2][288];   // 4 quarters * 72 (pad -> no bank conflicts)
    __shared__ float s_zero;

    const int tid  = threadIdx.x;
    const int w    = tid >> 5;
    const int lane = tid & 31;
    const int waveBase = blockIdx.x * 256 + w * 32;

    if (tid == 0) s_zero = 0.0f;

    float sh[NSPH];
    float rb[NRBF];
    const int e = waveBase + lane;
    if (e < P) {
        const float rx = r_ij[3 * e + 0];
        const float ry = r_ij[3 * e + 1];
        const float rz = r_ij[3 * e + 2];
        const float d  = sqrtf(rx * rx + ry * ry + rz * rz);
        const float dsafe = (d == 0.0f) ? 1.0f : d;
        const float inv = 1.0f / dsafe;
        const float x = rx * inv, y = ry * inv, z = rz * inv;

        // Bessel RBF: sqrt(2/5)*sin((n+1)*pi/5*d)/dsafe via Chebyshev recurrence
        // sin((k+1)t) = 2cos(t)sin(kt) - sin((k-1)t)  -> one sincos total.
        const float t = 0.6283185307179586f * d;  // pi/5 * d
        float s1, c1;
        __sincosf(t, &s1, &c1);
        const float pref = 0.6324555320336759f * inv;  // sqrt(2/5)/dsafe
        const float twoc = 2.0f * c1;
        float sk = s1, skm1 = 0.0f;
        #pragma unroll
        for (int n = 0; n < NRBF; ++n) {
            rb[n] = pref * sk;
            const float sk1 = twoc * sk - skm1;
            skm1 = sk; sk = sk1;
        }

        // scalar node-pair scale: embed[z[idx]]  (embed is (Z_MAX,1))
        const float xi = embed[zmap[idx_i[e]]];
        const float xj = embed[zmap[idx_j[e]]];
        const float scale = INV_SQRT_4PI * xi * xj;

        // spherical harmonics (L<=3), reference ordering
        const float x2 = x * x, y2 = y * y, z2 = z * z;
        const float x2z2 = x2 + z2;
        const float s3  = 1.7320508075688772f;
        const float s15 = 3.8729833462074170f;
        const float s5  = 2.2360679774997896f;
        const float s7  = 2.6457513110645907f;
        const float c30 = 1.0801234497346435f;  // sqrt(42)/6
        const float c32 = 1.6201851746019651f;  // sqrt(168)/8
        const float sh20 = s15 * x * z;
        const float sh24 = 0.5f * s15 * (z2 - x2);
        sh[0]  = 1.0f;
        sh[1]  = s3 * x;
        sh[2]  = s3 * y;
        sh[3]  = s3 * z;
        sh[4]  = sh20;
        sh[5]  = s15 * x * y;
        sh[6]  = s5 * (y2 - 0.5f * x2z2);
        sh[7]  = s15 * y * z;
        sh[8]  = sh24;
        sh[9]  = c30 * (sh20 * z + sh24 * x);
        sh[10] = s7 * sh20 * y;
        sh[11] = c32 * (4.0f * y2 - x2z2) * x;
        sh[12] = 0.5f * s7 * y * (2.0f * y2 - 3.0f * x2z2);
        sh[13] = c32 * z * (4.0f * y2 - x2z2);
        sh[14] = s7 * sh24 * y;
        sh[15] = c30 * (sh24 * z - sh20 * x);
        #pragma unroll
        for (int l = 0; l < NSPH; ++l) sh[l] *= scale;
    } else {
        #pragma unroll
        for (int l = 0; l < NSPH; ++l) sh[l] = 0.0f;
        #pragma unroll
        for (int n = 0; n < NRBF; ++n) rb[n] = 0.0f;
    }
    #pragma unroll
    for (int l = 0; l < NSPH; ++l) s_sph[w][lane][l] = sh[l];
    #pragma unroll
    for (int n = 0; n < NRBF; ++n) s_rbf[w][lane][n] = rb[n];

    __syncthreads();

    // Per-lane D-matrix role: N = lane&15, row half = lane>>4 (M or M+8).
    const int col   = lane & 15;
    const int half  = lane >> 4;
    const int which = col >> 3;      // 0 -> edge e0 (N 0-7), 1 -> edge e1 (N 8-15)
    const int nrb   = col & 7;
    const int whq   = which * 2 + half;          // staging quarter
    const int f0    = 4 * lane;                  // coalesced readback: edge e0 bytes
    const int f1    = 128 + 4 * lane;            // edge e1 bytes
    const int rd0   = (f0 >> 6) * 72 + (f0 & 63);
    const int rd1   = (f1 >> 6) * 72 + (f1 & 63);

    #pragma unroll 2
    for (int p = 0; p < 16; ++p) {
        const int e0 = 2 * p, e1 = 2 * p + 1;
        // A (16x4 f32): VGPR0 lanes0-15 = K0 = sph_e0[m]; VGPR1 lanes0-15 = K1 = sph_e1[m];
        // lanes16-31 (K2,K3) = 0. Zero-slot address select keeps EXEC all-1s.
        const float* pa0 = (lane < 16) ? &s_sph[w][e0][lane] : &s_zero;
        const float* pa1 = (lane < 16) ? &s_sph[w][e1][lane] : &s_zero;
        // B (4x16 f32): VGPR0 lanes0-15 = row K0 -> rbf_e0[n] (n<8) else 0;
        //               VGPR1 lanes0-15 = row K1 -> rbf_e1[n-8] for 8<=n<16;
        //               lanes16-31 (rows K2,K3) = 0.
        const float* pb0 = (lane < 8) ? &s_rbf[w][e0][lane] : &s_zero;
        const float* pb1 = (lane >= 8 && lane < 16) ? &s_rbf[w][e1][lane - 8] : &s_zero;

        v2f a, b;
        a.x = *pa0; a.y = *pa1;
        b.x = *pb0; b.y = *pb1;
        v8f c = {};
        // D[m,n<8] = sph_e0[m]*rbf_e0[n]; D[m,n>=8] = sph_e1[m]*rbf_e1[n-8]
        v8f dmat = __builtin_amdgcn_wmma_f32_16x16x4_f32(
            false, a, false, b, (short)0, c, false, false);

        // Scatter D into padded staging: dword addr = whq*72 + r*8 + nrb
        // (72 mod 64 = 8 -> quarters hit banks {0,8,16,24}+ofs: conflict-free).
        float* stg = &s_stage[w][p & 1][0];
        float* swr = stg + whq * 72 + nrb;
        #pragma unroll
        for (int r = 0; r < 8; ++r) swr[r * 8] = dmat[r];

        // wave-scope ordering: D scatter above must precede readback below
        __builtin_amdgcn_fence(__ATOMIC_ACQ_REL, "wavefront");

        const v4f o0 = *(const v4f*)(stg + rd0);
        const v4f o1 = *(const v4f*)(stg + rd1);

        const int eg0 = waveBase + e0;
        float* ob = out + (size_t)eg0 * 128;
        if (eg0 < P)                                  // wave-uniform branches:
            __builtin_nontemporal_store(o0, (v4f*)(ob + f0));   // 512B contiguous/wave
        if (eg0 + 1 < P)
            __builtin_nontemporal_store(o1, (v4f*)(ob + f1));   // 512B contiguous/wave
    }
}

extern "C" void kernel_launch(void* const* d_in, const int* in_sizes, int n_in,
                              void* d_out, int out_size, void* d_ws, size_t ws_size,
                              hipStream_t stream) {
    const int*   zmap  = (const int*)d_in[0];
    const int*   idx_i = (const int*)d_in[1];
    const int*   idx_j = (const int*)d_in[2];
    const float* r_ij  = (const float*)d_in[3];
    const float* embed = (const float*)d_in[4];
    float* out = (float*)d_out;
    const int P = in_sizes[1];

    const int blocks = (P + 255) / 256;
    edge_feature_kernel<<<blocks, 256, 0, stream>>>(zmap, idx_i, idx_j, r_ij,
                                                    embed, out, P);
}